// UnifiedQCNNQLSTM_65481071400632
// MI455X (gfx1250) — compile-verified
//
#include <hip/hip_runtime.h>
#include <math.h>

// ---------------------------------------------------------------------------
// UnifiedQCNNQLSTM on MI455X (gfx1250, wave32)
//  Stage 1: 8-qubit statevector sim, one sample per wave, state in VGPRs,
//           cross-lane qubits via shfl_xor (qubits 3..7 = lane bits).
//  Stage 2: sequential QLSTM scan, single resident wave, q staged in LDS.
//  Stage 3: classifier head 2048x4 @ 4x32 via V_WMMA_F32_16X16X4_F32 (+log_softmax).
// ---------------------------------------------------------------------------

typedef __attribute__((ext_vector_type(2))) float v2f;
typedef __attribute__((ext_vector_type(8))) float v8f;

#define PI_F 3.14159265358979323846f

__device__ __forceinline__ float2 cmulf2(float2 a, float2 b) {
    return make_float2(a.x * b.x - a.y * b.y, a.x * b.y + a.y * b.x);
}

// ---- 8-qubit (256-amp) wave-resident primitives -----------------------------
// amp index = lane*8 + r  (little-endian): qubit q<3 -> bit q of r,
// qubit q>=3 -> bit (q-3) of lane.

__device__ __forceinline__ void gate_diag8(float2 (&psi)[8], int q, int lane,
                                           float2 d0, float2 d1) {
#pragma unroll
    for (int r = 0; r < 8; ++r) {
        int bit = (q < 3) ? ((r >> q) & 1) : ((lane >> (q - 3)) & 1);
        float2 d = bit ? d1 : d0;
        psi[r] = cmulf2(psi[r], d);
    }
}

__device__ __forceinline__ void gate_real8(float2 (&psi)[8], int q, int lane,
                                           float m00, float m01, float m10, float m11) {
    if (q < 3) {
        const int m = 1 << q;
#pragma unroll
        for (int r = 0; r < 8; ++r) {
            if (!(r & m)) {
                float2 a0 = psi[r], a1 = psi[r | m];
                psi[r]     = make_float2(m00 * a0.x + m01 * a1.x, m00 * a0.y + m01 * a1.y);
                psi[r | m] = make_float2(m10 * a0.x + m11 * a1.x, m10 * a0.y + m11 * a1.y);
            }
        }
    } else {
        const int lm = 1 << (q - 3);
        const int hi = (lane >> (q - 3)) & 1;
        const float cown = hi ? m11 : m00;
        const float coth = hi ? m10 : m01;
#pragma unroll
        for (int r = 0; r < 8; ++r) {
            float ox = __shfl_xor(psi[r].x, lm, 32);
            float oy = __shfl_xor(psi[r].y, lm, 32);
            psi[r] = make_float2(cown * psi[r].x + coth * ox,
                                 cown * psi[r].y + coth * oy);
        }
    }
}

__device__ __forceinline__ void gate_cnot8(float2 (&psi)[8], int c, int t, int lane) {
    if (t < 3) {
        const int tm = 1 << t;
        float2 np[8];
#pragma unroll
        for (int r = 0; r < 8; ++r) {
            int cb = (c < 3) ? ((r >> c) & 1) : ((lane >> (c - 3)) & 1);
            np[r] = cb ? psi[r ^ tm] : psi[r];
        }
#pragma unroll
        for (int r = 0; r < 8; ++r) psi[r] = np[r];
    } else {
        const int lm = 1 << (t - 3);
#pragma unroll
        for (int r = 0; r < 8; ++r) {
            float ox = __shfl_xor(psi[r].x, lm, 32);
            float oy = __shfl_xor(psi[r].y, lm, 32);
            int cb = (c < 3) ? ((r >> c) & 1) : ((lane >> (c - 3)) & 1);
            if (cb) psi[r] = make_float2(ox, oy);
        }
    }
}

__device__ __forceinline__ void gate_rz8(float2 (&psi)[8], int q, int lane, float t) {
    float s, c;
    __sincosf(0.5f * t, &s, &c);
    gate_diag8(psi, q, lane, make_float2(c, -s), make_float2(c, s));
}
__device__ __forceinline__ void gate_ry8(float2 (&psi)[8], int q, int lane, float t) {
    float s, c;
    __sincosf(0.5f * t, &s, &c);
    gate_real8(psi, q, lane, c, -s, s, c);
}
__device__ __forceinline__ void gate_ph8(float2 (&psi)[8], int q, int lane, float t) {
    float s, c;
    __sincosf(t, &s, &c);
    gate_diag8(psi, q, lane, make_float2(1.f, 0.f), make_float2(c, s));
}
__device__ __forceinline__ void gate_h8(float2 (&psi)[8], int q, int lane) {
    const float R = 0.70710678118654752f;
    gate_real8(psi, q, lane, R, R, R, -R);
}

__device__ __forceinline__ void conv_block8(float2 (&psi)[8], int a, int b, int lane,
                                            const float* __restrict__ p) {
    gate_rz8(psi, b, lane, -0.5f * PI_F);
    gate_cnot8(psi, b, a, lane);
    gate_rz8(psi, a, lane, p[0]);
    gate_ry8(psi, b, lane, p[1]);
    gate_cnot8(psi, a, b, lane);
    gate_ry8(psi, b, lane, p[2]);
    gate_cnot8(psi, b, a, lane);
    gate_rz8(psi, a, lane, 0.5f * PI_F);
}

__device__ __forceinline__ void pool_block8(float2 (&psi)[8], int a, int b, int lane,
                                            const float* __restrict__ p) {
    gate_rz8(psi, b, lane, -0.5f * PI_F);
    gate_cnot8(psi, b, a, lane);
    gate_rz8(psi, a, lane, p[0]);
    gate_ry8(psi, b, lane, p[1]);
    gate_cnot8(psi, a, b, lane);
    gate_ry8(psi, b, lane, p[2]);
}

// ---- Stage 1: QCNN, one sample per wave -------------------------------------

__global__ __launch_bounds__(256) void qcnn_kernel(const float* __restrict__ sent,
                                                   const float* __restrict__ w,
                                                   float* __restrict__ qout) {
    const int lane = threadIdx.x & 31;
    const int s    = blockIdx.x * (blockDim.x >> 5) + (threadIdx.x >> 5);  // 0..2047
    const float* x = sent + s * 8;

    float2 psi[8];
#pragma unroll
    for (int r = 0; r < 8; ++r) psi[r] = make_float2(0.f, 0.f);
    if (lane == 0) psi[0].x = 1.0f;

    // ZFeatureMap(reps=1): H then P(2*x_q) per qubit
#pragma unroll
    for (int q = 0; q < 8; ++q) {
        gate_h8(psi, q, lane);
        gate_ph8(psi, q, lane, 2.0f * x[q]);
    }

    int k = 0;
    {
        constexpr int C1[8][2] = {{0,1},{2,3},{4,5},{6,7},{1,2},{3,4},{5,6},{7,0}};
#pragma unroll
        for (int i = 0; i < 8; ++i) { conv_block8(psi, C1[i][0], C1[i][1], lane, w + k); k += 3; }
    }
    {
        constexpr int P1[4][2] = {{0,4},{1,5},{2,6},{3,7}};
#pragma unroll
        for (int i = 0; i < 4; ++i) { pool_block8(psi, P1[i][0], P1[i][1], lane, w + k); k += 3; }
    }
    {
        constexpr int C2[4][2] = {{0,1},{2,3},{1,2},{3,0}};
#pragma unroll
        for (int i = 0; i < 4; ++i) { conv_block8(psi, C2[i][0], C2[i][1], lane, w + k); k += 3; }
    }
    {
        constexpr int P2[2][2] = {{0,2},{1,3}};
#pragma unroll
        for (int i = 0; i < 2; ++i) { pool_block8(psi, P2[i][0], P2[i][1], lane, w + k); k += 3; }
    }
    {
        constexpr int C3[2][2] = {{0,1},{1,0}};
#pragma unroll
        for (int i = 0; i < 2; ++i) { conv_block8(psi, C3[i][0], C3[i][1], lane, w + k); k += 3; }
    }
    pool_block8(psi, 0, 1, lane, w + k);  // POOL3

    // <Z_7>: bit7 of amp index == bit4 of lane
    float part = 0.f;
#pragma unroll
    for (int r = 0; r < 8; ++r) part += psi[r].x * psi[r].x + psi[r].y * psi[r].y;
    part = (lane < 16) ? part : -part;
#pragma unroll
    for (int m = 16; m >= 1; m >>= 1) part += __shfl_xor(part, m, 32);
    if (lane == 0) qout[s] = part;
}

// ---- Stage 2: QLSTM scan, single resident wave ------------------------------

__global__ __launch_bounds__(32) void qlstm_kernel(const float* __restrict__ qin,
        const float* __restrict__ Wf, const float* __restrict__ bf,
        const float* __restrict__ Wi, const float* __restrict__ bi,
        const float* __restrict__ Wu, const float* __restrict__ bu,
        const float* __restrict__ Wo, const float* __restrict__ bo,
        const float* __restrict__ thf, const float* __restrict__ thi,
        const float* __restrict__ thu, const float* __restrict__ tho,
        float* __restrict__ hs) {
    __shared__ float qs[2048];
    const int lane = threadIdx.x & 31;
    for (int i = lane; i < 2048; i += 32) qs[i] = qin[i];
    __syncthreads();

    const int g = lane & 3;  // each lane simulates gate circuit g (replicated x8)
    const float* Wg = (g == 0) ? Wf : (g == 1) ? Wi : (g == 2) ? Wu : Wo;
    const float* bg = (g == 0) ? bf : (g == 1) ? bi : (g == 2) ? bu : bo;
    const float* tg = (g == 0) ? thf : (g == 1) ? thi : (g == 2) ? thu : tho;

    float Wr[20], bth[4];
#pragma unroll
    for (int j = 0; j < 20; ++j) Wr[j] = Wg[j];
#pragma unroll
    for (int j = 0; j < 4; ++j) bth[j] = bg[j] + tg[j];

    float h[4] = {0.f, 0.f, 0.f, 0.f};
    float c[4] = {0.f, 0.f, 0.f, 0.f};

    for (int s = 0; s < 2048; ++s) {
        const float qv = qs[s];
        float ang[4];
#pragma unroll
        for (int j = 0; j < 4; ++j)
            ang[j] = Wr[j * 5 + 0] * qv + Wr[j * 5 + 1] * h[0] + Wr[j * 5 + 2] * h[1] +
                     Wr[j * 5 + 3] * h[2] + Wr[j * 5 + 4] * h[3] + bth[j];

        // 4-qubit circuit: RX(ang_wq) each wire, CNOT ring, <Z_w> per wire
        float2 a16[16];
#pragma unroll
        for (int i = 0; i < 16; ++i) a16[i] = make_float2(0.f, 0.f);
        a16[0].x = 1.0f;
#pragma unroll
        for (int wq = 0; wq < 4; ++wq) {
            float sn, cs;
            __sincosf(0.5f * ang[wq], &sn, &cs);
            const int m = 1 << wq;
#pragma unroll
            for (int i = 0; i < 16; ++i) {
                if (!(i & m)) {
                    float2 a0 = a16[i], a1 = a16[i | m];
                    // RX: [[c, -i s],[-i s, c]]
                    a16[i]     = make_float2(cs * a0.x + sn * a1.y, cs * a0.y - sn * a1.x);
                    a16[i | m] = make_float2(sn * a0.y + cs * a1.x, -sn * a0.x + cs * a1.y);
                }
            }
        }
        {
            constexpr int CT[4][2] = {{0,1},{1,2},{2,3},{3,0}};
#pragma unroll
            for (int e = 0; e < 4; ++e) {
                const int cm = 1 << CT[e][0], tm = 1 << CT[e][1];
                float2 np[16];
#pragma unroll
                for (int i = 0; i < 16; ++i) np[i] = (i & cm) ? a16[i ^ tm] : a16[i];
#pragma unroll
                for (int i = 0; i < 16; ++i) a16[i] = np[i];
            }
        }
        float z[4] = {0.f, 0.f, 0.f, 0.f};
#pragma unroll
        for (int i = 0; i < 16; ++i) {
            float p = a16[i].x * a16[i].x + a16[i].y * a16[i].y;
#pragma unroll
            for (int wq = 0; wq < 4; ++wq) z[wq] += ((i >> wq) & 1) ? -p : p;
        }

        // transpose z across lanes 0..3 -> every lane gets the full 4x4
        float zm[4][4];
#pragma unroll
        for (int j = 0; j < 4; ++j) {
            zm[0][j] = __shfl(z[j], 0, 32);
            zm[1][j] = __shfl(z[j], 1, 32);
            zm[2][j] = __shfl(z[j], 2, 32);
            zm[3][j] = __shfl(z[j], 3, 32);
        }
#pragma unroll
        for (int j = 0; j < 4; ++j) {
            float fj = 1.f / (1.f + __expf(-zm[0][j]));
            float ij = 1.f / (1.f + __expf(-zm[1][j]));
            float uj = tanhf(zm[2][j]);
            float oj = 1.f / (1.f + __expf(-zm[3][j]));
            c[j] = fj * c[j] + ij * uj;
            h[j] = oj * tanhf(c[j]);
        }
        if (lane == 0) {
            hs[s * 4 + 0] = h[0];
            hs[s * 4 + 1] = h[1];
            hs[s * 4 + 2] = h[2];
            hs[s * 4 + 3] = h[3];
        }
    }
}

// ---- Stage 3: classifier head via V_WMMA_F32_16X16X4_F32 --------------------
// logits = hs(2048x4) @ Wt^T(4x32) + bt, then row log_softmax.
// One wave per 16-row tile; two WMMAs cover N=32.

__global__ __launch_bounds__(32) void head_kernel(const float* __restrict__ hs,
                                                  const float* __restrict__ Wt,
                                                  const float* __restrict__ bt,
                                                  float* __restrict__ out) {
    const int lane = threadIdx.x & 31;
    const int rt   = blockIdx.x;            // row tile 0..127
    const int n16  = lane & 15;
    const int kh   = (lane < 16) ? 0 : 2;   // ISA f32 A/B layout: lanes16-31 -> K=2,3

    // A (16x4): lane L holds A[M=L%16][K=kh..kh+1]
    v2f a;
    a.x = hs[(rt * 16 + n16) * 4 + kh + 0];
    a.y = hs[(rt * 16 + n16) * 4 + kh + 1];
    // B (4x16): lane L holds B[K=kh..kh+1][N=L%16]; B[k][n] = Wt[n][k]
    v2f b0, b1;
    b0.x = Wt[n16 * 4 + kh + 0];
    b0.y = Wt[n16 * 4 + kh + 1];
    b1.x = Wt[(16 + n16) * 4 + kh + 0];
    b1.y = Wt[(16 + n16) * 4 + kh + 1];

    v8f cz = {0.f, 0.f, 0.f, 0.f, 0.f, 0.f, 0.f, 0.f};
    v8f d0 = __builtin_amdgcn_wmma_f32_16x16x4_f32(false, a, false, b0, (short)0, cz, false, false);
    v8f d1 = __builtin_amdgcn_wmma_f32_16x16x4_f32(false, a, false, b1, (short)0, cz, false, false);

    const float bb0 = bt[n16];
    const float bb1 = bt[16 + n16];
#pragma unroll
    for (int r = 0; r < 8; ++r) { d0[r] += bb0; d1[r] += bb1; }

    // log_softmax per row: D layout -> reg r, lanes<16: row r cols 0..15,
    // lanes>=16: row r+8. xor masks 1..8 stay within each 16-lane half.
#pragma unroll
    for (int r = 0; r < 8; ++r) {
        float m = fmaxf(d0[r], d1[r]);
#pragma unroll
        for (int msk = 8; msk >= 1; msk >>= 1) m = fmaxf(m, __shfl_xor(m, msk, 32));
        float e = __expf(d0[r] - m) + __expf(d1[r] - m);
#pragma unroll
        for (int msk = 8; msk >= 1; msk >>= 1) e += __shfl_xor(e, msk, 32);
        const float lse = m + __logf(e);
        const int row = rt * 16 + r + ((lane < 16) ? 0 : 8);
        out[row * 32 + n16]      = d0[r] - lse;
        out[row * 32 + 16 + n16] = d1[r] - lse;
    }
}

// ---------------------------------------------------------------------------

extern "C" void kernel_launch(void* const* d_in, const int* in_sizes, int n_in,
                              void* d_out, int out_size, void* d_ws, size_t ws_size,
                              hipStream_t stream) {
    (void)in_sizes; (void)n_in; (void)out_size; (void)ws_size;
    const float* sent = (const float*)d_in[0];
    const float* qw   = (const float*)d_in[1];
    const float* Wf   = (const float*)d_in[2];
    const float* bf   = (const float*)d_in[3];
    const float* Wi   = (const float*)d_in[4];
    const float* bi   = (const float*)d_in[5];
    const float* Wu   = (const float*)d_in[6];
    const float* bu   = (const float*)d_in[7];
    const float* Wo   = (const float*)d_in[8];
    const float* bo   = (const float*)d_in[9];
    const float* thf  = (const float*)d_in[10];
    const float* thi  = (const float*)d_in[11];
    const float* thu  = (const float*)d_in[12];
    const float* tho  = (const float*)d_in[13];
    const float* Wt   = (const float*)d_in[14];
    const float* bt   = (const float*)d_in[15];
    float* out = (float*)d_out;

    float* qbuf  = (float*)d_ws;     // 2048 floats
    float* hsbuf = qbuf + 2048;      // 2048*4 floats

    qcnn_kernel<<<256, 256, 0, stream>>>(sent, qw, qbuf);
    qlstm_kernel<<<1, 32, 0, stream>>>(qbuf, Wf, bf, Wi, bi, Wu, bu, Wo, bo,
                                       thf, thi, thu, tho, hsbuf);
    head_kernel<<<128, 32, 0, stream>>>(hsbuf, Wt, bt, out);
}